// TimesNet_90632399880410
// MI455X (gfx1250) — compile-verified
//
#include <hip/hip_runtime.h>
#include <hip/hip_bf16.h>

// PeriodicityTransform on gfx1250.
// One workgroup (512 threads = 16 wave32) per series (BN = 32*64 = 2048).
// FFT-4096 via four-step: two batches of 64-point DFTs expressed as 64x64x64
// f32 matmuls on V_WMMA_F32_16X16X4_F32 (full f32 precision so top-k spectral
// ranking matches the f32 reference).

typedef float v2f __attribute__((ext_vector_type(2)));
typedef float v8f __attribute__((ext_vector_type(8)));

#define T_LEN   4096
#define N_SER   64
#define B_BATCH 32
#define BN      (B_BATCH * N_SER)
#define KP      5
#define NTHREADS 512

__device__ __forceinline__ v8f wmma4(v2f a, v2f b, v8f c) {
  // V_WMMA_F32_16X16X4_F32: D(16x16,f32) = A(16x4,f32) x B(4x16,f32) + C
  // 8-arg pattern per probe doc: (neg_a, A, neg_b, B, c_mod, C, reuse_a, reuse_b)
  return __builtin_amdgcn_wmma_f32_16x16x4_f32(false, a, false, b, (short)0, c,
                                               false, false);
}

__global__ __launch_bounds__(NTHREADS)
void periodicity_kernel(const float* __restrict__ x, float* __restrict__ out) {
  __shared__ float sSeq[T_LEN];      // raw series, also A-matrix of stage 1
  __shared__ float sAr[T_LEN];       // stage1 result Re; later aliased as mag2
  __shared__ float sAi[T_LEN];       // stage1 result Im
  __shared__ float sTab[64];         // cos(2*pi*m/64) table (Re & Im of W64)
  __shared__ float rV[NTHREADS];
  __shared__ int   rI[NTHREADS];
  __shared__ int   sKidx[KP];
  __shared__ int   sStart[KP];

  const int tid  = threadIdx.x;
  const int bn   = blockIdx.x;              // series id in [0, 2048)
  const int lane = tid & 31;
  const int wave = tid >> 5;                // [0,16)
  const int l15  = lane & 15;
  const int hi2  = (lane >> 4) << 1;        // 0 or 2 : K sub-offset of fragment
  const int vrow = (lane >> 4) << 3;        // 0 or 8 : C/D row sub-offset

  // ---- load series: seq[t] = x[b][t][n], bn = b*64 + n ----
  {
    const int b = bn >> 6, n = bn & 63;
    const float* xp = x + (size_t)b * T_LEN * N_SER + n;
    for (int t = tid; t < T_LEN; t += NTHREADS)
      sSeq[t] = xp[(size_t)t * N_SER];
  }
  if (tid < 64) sTab[tid] = cospif((float)tid / 32.0f); // cos(2*pi*tid/64)
  __syncthreads();

  // W64[j][k] = e^{-2*pi*i*jk/64}: Re = tab[(jk)&63], Im = -tab[((jk)+48)&63]
  // (since -sin(th) = -cos(th - pi/2) and pi/2 <-> m=16, -16 = +48 mod 64)

  // ---- stage 1: A[n1][k2] = sum_n2 M[n1][n2] * W64[n2][k2],
  //      M[n1][n2] = seq[n1 + 64*n2]. 16 output tiles; wave w -> tile w. ----
  {
    const int tm = wave >> 2, tn = wave & 3;
    const int arow = tm * 16 + l15;          // n1
    const int bcol = tn * 16 + l15;          // k2
    v8f ar = {}, ai = {};
    for (int k0 = 0; k0 < 64; k0 += 4) {
      const int ka = k0 + hi2;               // n2 slice for this half-wave
      v2f a;                                  // A fragment (16x4 layout)
      a.x = sSeq[arow + 64 * ka];
      a.y = sSeq[arow + 64 * (ka + 1)];
      const int m0 = (ka * bcol) & 63;
      const int m1 = ((ka + 1) * bcol) & 63;
      v2f br, bi;                             // B fragment (4x16 layout)
      br.x = sTab[m0];               br.y = sTab[m1];
      bi.x = -sTab[(m0 + 48) & 63];  bi.y = -sTab[(m1 + 48) & 63];
      ar = wmma4(a, br, ar);
      ai = wmma4(a, bi, ai);
    }
    for (int v = 0; v < 8; ++v) {            // C/D layout store
      const int row = tm * 16 + v + vrow;
      const int col = tn * 16 + l15;
      sAr[row * 64 + col] = ar[v];
      sAi[row * 64 + col] = ai[v];
    }
  }
  __syncthreads();

  // ---- twiddle: B[n1][k2] = A[n1][k2] * e^{-2*pi*i*n1*k2/4096} ----
  for (int i = tid; i < T_LEN; i += NTHREADS) {
    const int n1 = i >> 6, k2 = i & 63;
    float s, c;
    sincospif(-(float)(n1 * k2) / 2048.0f, &s, &c);
    const float ar = sAr[i], ai = sAi[i];
    sAr[i] = ar * c - ai * s;
    sAi[i] = ar * s + ai * c;
  }
  __syncthreads();

  // ---- stage 2: X[k1][k2] = sum_n1 W64[k1][n1] * B[n1][k2];
  //      mag2 at linear index k = 64*k1 + k2 (== frequency index). ----
  {
    const int tm = wave >> 2, tn = wave & 3;
    const int arow = tm * 16 + l15;          // k1
    const int bcol = tn * 16 + l15;          // k2
    v8f xr1 = {}, xr2 = {}, xi = {};
    for (int k0 = 0; k0 < 64; k0 += 4) {
      const int ka = k0 + hi2;               // n1
      const int m0 = (arow * ka) & 63;
      const int m1 = (arow * (ka + 1)) & 63;
      v2f wr, wi;
      wr.x = sTab[m0];               wr.y = sTab[m1];
      wi.x = -sTab[(m0 + 48) & 63];  wi.y = -sTab[(m1 + 48) & 63];
      v2f br, bi;
      br.x = sAr[ka * 64 + bcol];    br.y = sAr[(ka + 1) * 64 + bcol];
      bi.x = sAi[ka * 64 + bcol];    bi.y = sAi[(ka + 1) * 64 + bcol];
      xr1 = wmma4(wr, br, xr1);      // Re part 1
      xr2 = wmma4(wi, bi, xr2);      // Re part 2 (subtract later; f32 WMMA
                                     // has no A/B negate, only C-negate)
      xi  = wmma4(wr, bi, xi);       // Im, chained accumulate
      xi  = wmma4(wi, br, xi);
    }
    __syncthreads();                 // all reads of sAr/sAi done -> alias as mag
    float* sMag = sAr;
    for (int v = 0; v < 8; ++v) {
      const float XR = xr1[v] - xr2[v];
      const float XI = xi[v];
      const int row = tm * 16 + v + vrow;
      const int col = tn * 16 + l15;
      sMag[row * 64 + col] = XR * XR + XI * XI;
    }
  }
  __syncthreads();

  // ---- top-5 over k in [1, 2048] (DC excluded, ties -> lower index) ----
  float* sMag = sAr;
  for (int iter = 0; iter < KP; ++iter) {
    float best = -1.0f; int bidx = 0x7fffffff;
    for (int k = 1 + tid; k <= 2048; k += NTHREADS) {
      const float m = sMag[k];
      if (m > best || (m == best && k < bidx)) { best = m; bidx = k; }
    }
    rV[tid] = best; rI[tid] = bidx;
    __syncthreads();
    for (int s = NTHREADS / 2; s > 0; s >>= 1) {
      if (tid < s) {
        const float v2 = rV[tid + s]; const int i2 = rI[tid + s];
        if (v2 > rV[tid] || (v2 == rV[tid] && i2 < rI[tid])) {
          rV[tid] = v2; rI[tid] = i2;
        }
      }
      __syncthreads();
    }
    if (tid == 0) { sKidx[iter] = rI[0]; sMag[rI[0]] = -1.0f; }
    __syncthreads();
  }

  // ---- period / cycle integer math + scalar outputs ----
  if (tid < KP) {
    const int kI = sKidx[tid];                    // >= 1 by construction
    int p = T_LEN / kI;
    p = p > 512 ? 512 : p;                        // min(p, PMAX)
    p = p < 1 ? 1 : p;                            // max(p, min_thresh=1)
    int c = T_LEN / p; c = c < 1 ? 1 : c;
    sStart[tid] = T_LEN - c * p;
    const size_t V = (size_t)BN * KP * T_LEN;
    float* outs = out + V;
    outs[(size_t)bn * KP + tid]                    = (float)c;  // cycles
    outs[(size_t)BN * KP + (size_t)bn * KP + tid]  = (float)p;  // periods
  }
  __syncthreads();

  // ---- dense masked values: out[bn,k,t] = (t >= start) ? seq[t] : 0 ----
  for (int kk = 0; kk < KP; ++kk) {
    const int st = sStart[kk];
    const size_t base = ((size_t)bn * KP + kk) * T_LEN;
    for (int t = tid; t < T_LEN; t += NTHREADS)
      out[base + t] = (t >= st) ? sSeq[t] : 0.0f;
  }
}

extern "C" void kernel_launch(void* const* d_in, const int* in_sizes, int n_in,
                              void* d_out, int out_size, void* d_ws, size_t ws_size,
                              hipStream_t stream) {
  (void)in_sizes; (void)n_in; (void)out_size; (void)d_ws; (void)ws_size;
  const float* x = (const float*)d_in[0];
  float* out = (float*)d_out;
  periodicity_kernel<<<BN, NTHREADS, 0, stream>>>(x, out);
}